// NystromAttention_35313221107731
// MI455X (gfx1250) — compile-verified
//
#include <hip/hip_runtime.h>
#include <hip/hip_bf16.h>

// Problem constants (match reference)
constexpr int B = 4;
constexpr int N = 4096;
constexpr int D = 64;
constexpr int M = 64;

typedef float v2f __attribute__((ext_vector_type(2)));
typedef float v8f __attribute__((ext_vector_type(8)));

// ---------------- workspace layout (float offsets) ----------------
constexpr size_t OFF_QT   = 0;
constexpr size_t OFF_KT   = OFF_QT + (size_t)B*M*D;
constexpr size_t OFF_K2   = OFF_KT + (size_t)B*M*D;
constexpr size_t OFF_CS   = OFF_K2 + (size_t)B*M*M;
constexpr size_t OFF_PINV = OFF_CS + (size_t)B*M;
constexpr size_t OFF_T1   = OFF_PINV + (size_t)B*M*M;
constexpr size_t OFF_K3   = OFF_T1 + (size_t)B*N*M;

// ---------------- 1) gather landmark rows ----------------
__global__ __launch_bounds__(64) void gather_kernel(
    const float* __restrict__ Q, const float* __restrict__ K,
    const int* __restrict__ iq, const int* __restrict__ ik,
    float* __restrict__ QT, float* __restrict__ KT) {
  int b = blockIdx.y, m = blockIdx.x, d = threadIdx.x;
  QT[((size_t)b*M + m)*D + d] = Q[((size_t)b*N + iq[m])*D + d];
  KT[((size_t)b*M + m)*D + d] = K[((size_t)b*N + ik[m])*D + d];
}

// ---------------- 2) kernel2 = softmax(Qt @ Kt^T) + column sums ----------------
__global__ __launch_bounds__(64) void k2_kernel(
    const float* __restrict__ QT, const float* __restrict__ KT,
    float* __restrict__ K2, float* __restrict__ CS) {
  __shared__ float Kt[M*D];
  __shared__ float Qt[M*D];
  __shared__ float k2[M*M];
  int b = blockIdx.x, m = threadIdx.x;  // 64 threads, thread owns row m
  for (int i = m; i < M*D; i += 64) {
    Kt[i] = KT[(size_t)b*M*D + i];
    Qt[i] = QT[(size_t)b*M*D + i];
  }
  __syncthreads();
  float s[M];
  float smax = -1e30f;
  for (int l = 0; l < M; ++l) {
    float t = 0.f;
    for (int d = 0; d < D; ++d) t += Qt[m*D + d] * Kt[l*D + d];
    s[l] = t;
    smax = fmaxf(smax, t);
  }
  float ssum = 0.f;
  for (int l = 0; l < M; ++l) { s[l] = __expf(s[l] - smax); ssum += s[l]; }
  float inv = 1.f / ssum;
  for (int l = 0; l < M; ++l) {
    float p = s[l] * inv;
    k2[m*M + l] = p;
    K2[(size_t)b*M*M + m*M + l] = p;
  }
  __syncthreads();
  // column sum over rows (axis=-2)
  float c = 0.f;
  for (int i = 0; i < M; ++i) c += k2[i*M + m];
  CS[b*M + m] = c;
}

// ---------------- 3) Newton-Schulz pseudo-inverse of kernel2 (per batch) ----------------
__global__ __launch_bounds__(256) void inv_kernel(
    const float* __restrict__ K2, const float* __restrict__ CS,
    float* __restrict__ PINV) {
  __shared__ float mat[M*M];
  __shared__ float V[M*M];
  __shared__ float KV[M*M];
  __shared__ float P[M*M];
  int b = blockIdx.x, tid = threadIdx.x;

  // global scalar: max over all batches of column sums
  float mx = CS[0];
  for (int i = 1; i < B*M; ++i) mx = fmaxf(mx, CS[i]);
  float invmx = 1.f / mx;

  for (int t = 0; t < 16; ++t) {
    int flat = tid + t*256;
    mat[flat] = K2[(size_t)b*M*M + flat];
  }
  __syncthreads();
  // V = mat^T / mx
  for (int t = 0; t < 16; ++t) {
    int flat = tid + t*256;
    int i = flat >> 6, j = flat & 63;
    V[flat] = mat[j*M + i] * invmx;
  }
  __syncthreads();

  float r[16];
  for (int it = 0; it < 6; ++it) {
    // KV = mat @ V
    for (int t = 0; t < 16; ++t) {
      int flat = tid + t*256;
      int i = flat >> 6, j = flat & 63;
      float s = 0.f;
      for (int k = 0; k < M; ++k) s += mat[i*M + k] * V[k*M + j];
      r[t] = s;
    }
    __syncthreads();
    for (int t = 0; t < 16; ++t) KV[tid + t*256] = r[t];
    __syncthreads();
    // P = KV @ (7I - KV) = 7*KV - KV@KV
    for (int t = 0; t < 16; ++t) {
      int flat = tid + t*256;
      int i = flat >> 6, j = flat & 63;
      float s = 0.f;
      for (int k = 0; k < M; ++k) s += KV[i*M + k] * KV[k*M + j];
      r[t] = 7.f * KV[flat] - s;
    }
    __syncthreads();
    for (int t = 0; t < 16; ++t) P[tid + t*256] = r[t];
    __syncthreads();
    // P <- KV @ (15I - P) = 15*KV - KV@P   (register-staged, then overwrite P)
    for (int t = 0; t < 16; ++t) {
      int flat = tid + t*256;
      int i = flat >> 6, j = flat & 63;
      float s = 0.f;
      for (int k = 0; k < M; ++k) s += KV[i*M + k] * P[k*M + j];
      r[t] = 15.f * KV[flat] - s;
    }
    __syncthreads();
    for (int t = 0; t < 16; ++t) P[tid + t*256] = r[t];
    __syncthreads();
    // V <- 0.25 * V @ (13I - P) = 0.25*(13*V - V@P)
    for (int t = 0; t < 16; ++t) {
      int flat = tid + t*256;
      int i = flat >> 6, j = flat & 63;
      float s = 0.f;
      for (int k = 0; k < M; ++k) s += V[i*M + k] * P[k*M + j];
      r[t] = 0.25f * (13.f * V[flat] - s);
    }
    __syncthreads();
    for (int t = 0; t < 16; ++t) V[tid + t*256] = r[t];
    __syncthreads();
  }
  for (int t = 0; t < 16; ++t) {
    int flat = tid + t*256;
    PINV[(size_t)b*M*M + flat] = V[flat];
  }
}

// ---------------- 4) T1 = softmax(Q @ Kt^T) @ pinv  -> (B, N, M) ----------------
__global__ __launch_bounds__(256) void k1t1_kernel(
    const float* __restrict__ Q, const float* __restrict__ KT,
    const float* __restrict__ PINV, float* __restrict__ T1) {
  __shared__ float Kt[M*D];
  __shared__ float Pv[M*M];
  __shared__ float p[4][M];
  int b = blockIdx.y, tid = threadIdx.x;
  for (int i = tid; i < M*D; i += 256) {
    Kt[i] = KT[(size_t)b*M*D + i];
    Pv[i] = PINV[(size_t)b*M*M + i];
  }
  __syncthreads();
  int g = tid >> 6, m = tid & 63;          // 4 groups of 64 lanes; group owns row n
  int n = blockIdx.x * 4 + g;
  const float* qrow = Q + ((size_t)b*N + n)*D;
  float s = 0.f;
  for (int d = 0; d < D; ++d) s += qrow[d] * Kt[m*D + d];
  p[g][m] = s;
  __syncthreads();
  float mx = p[g][0];
  for (int l = 1; l < M; ++l) mx = fmaxf(mx, p[g][l]);
  float e = __expf(s - mx);
  __syncthreads();
  p[g][m] = e;
  __syncthreads();
  float sum = 0.f;
  for (int l = 0; l < M; ++l) sum += p[g][l];
  float prob = e / sum;
  __syncthreads();
  p[g][m] = prob;
  __syncthreads();
  // thread m plays role of output column d: T1[b,n,d] = sum_l p[l] * pinv[l][d]
  float t = 0.f;
  for (int l = 0; l < M; ++l) t += p[g][l] * Pv[l*M + m];
  T1[((size_t)b*N + n)*M + m] = t;
}

// ---------------- 5) kernel3 = softmax over N of (Qt @ K^T) -> (B, M, N) ----------------
__global__ __launch_bounds__(256) void k3_kernel(
    const float* __restrict__ QT, const float* __restrict__ K,
    float* __restrict__ K3) {
  __shared__ float q[D];
  __shared__ float red[256];
  int b = blockIdx.y, m = blockIdx.x, tid = threadIdx.x;
  if (tid < D) q[tid] = QT[((size_t)b*M + m)*D + tid];
  __syncthreads();
  float sc[16];
  float lmax = -1e30f;
  for (int i = 0; i < 16; ++i) {
    int l = tid + i*256;
    const float* kr = K + ((size_t)b*N + l)*D;
    float s = 0.f;
    for (int d = 0; d < D; ++d) s += q[d] * kr[d];
    sc[i] = s;
    lmax = fmaxf(lmax, s);
  }
  red[tid] = lmax;
  __syncthreads();
  for (int off = 128; off; off >>= 1) {
    if (tid < off) red[tid] = fmaxf(red[tid], red[tid + off]);
    __syncthreads();
  }
  float mx = red[0];
  __syncthreads();
  float lsum = 0.f;
  for (int i = 0; i < 16; ++i) { sc[i] = __expf(sc[i] - mx); lsum += sc[i]; }
  red[tid] = lsum;
  __syncthreads();
  for (int off = 128; off; off >>= 1) {
    if (tid < off) red[tid] += red[tid + off];
    __syncthreads();
  }
  float inv = 1.f / red[0];
  for (int i = 0; i < 16; ++i)
    K3[(size_t)b*M*N + (size_t)m*N + tid + i*256] = sc[i] * inv;
}

// ---------------- 6) out = T1 @ K3  via V_WMMA_F32_16X16X4_F32 ----------------
// 128x128 output tile per 256-thread (8-wave) workgroup; K = 64.
// Both LDS tiles use an XOR swizzle on the K index so that WMMA fragment
// loads (16 lanes reading rows/cols 0..15 at the same K) sweep all 64 LDS
// banks conflict-free without padding: element (row,k) -> row*64 + (k ^ 4*(row&15)).
__global__ __launch_bounds__(256) void gemm_wmma_kernel(
    const float* __restrict__ T1, const float* __restrict__ K3,
    float* __restrict__ out) {
  __shared__ alignas(16) float Asb[128*64];   // A tile, swizzled: (row, k)
  __shared__ alignas(16) float Btb[128*64];   // B tile, TRANSPOSED + swizzled: (col, k)
  int b  = blockIdx.z;
  int i0 = blockIdx.y * 128;
  int j0 = blockIdx.x * 128;
  int tid = threadIdx.x;

  const float* Ag = T1 + ((size_t)b*N + i0) * M;          // row-major, ld=64
  const float* Bg = K3 + (size_t)b*M*N + j0;              // row-major, ld=N

  // stage A tile: 128 rows x 64 k, float4 global loads, swizzled float4 LDS stores
  #pragma unroll
  for (int r = 0; r < 8; ++r) {
    int flat = tid + r*256;           // float4 index
    int row  = flat >> 4;             // 16 float4 per row
    int c4   = flat & 15;
    float4 v = ((const float4*)(Ag + (size_t)row*M))[c4];
    int sc4  = ((c4*4) ^ (4*(row & 15)));   // 4-aligned, contiguity preserved
    *((float4*)(Asb + row*64 + sc4)) = v;
  }
  // stage B tile transposed: (col, k). Coalesced scalar global loads
  // (128 consecutive cols per instruction), vector LDS stores.
  #pragma unroll
  for (int r = 0; r < 8; ++r) {
    int flat = tid + r*256;           // 2048 tasks: 128 cols x 16 k-groups
    int col  = flat & 127;
    int k4   = flat >> 7;
    float4 v;
    v.x = Bg[(size_t)(k4*4 + 0)*N + col];
    v.y = Bg[(size_t)(k4*4 + 1)*N + col];
    v.z = Bg[(size_t)(k4*4 + 2)*N + col];
    v.w = Bg[(size_t)(k4*4 + 3)*N + col];
    int sk4 = ((k4*4) ^ (4*(col & 15)));
    *((float4*)(Btb + col*64 + sk4)) = v;
  }
  __syncthreads();

  int w = tid >> 5, lane = tid & 31;
  int wr = w >> 2, wc = w & 3;        // waves: 2 (row) x 4 (col); wave tile 64x32
  int lrow = lane & 15;
  int lk   = (lane >> 4) << 1;        // lanes 0-15 -> K{0,1}; lanes 16-31 -> K{2,3}
  int swz  = 4 * (lrow & 15);         // per-lane XOR key (rows/cols 0..15)

  v8f acc[4][2];
  #pragma unroll
  for (int rt = 0; rt < 4; ++rt)
    #pragma unroll
    for (int ct = 0; ct < 2; ++ct)
      #pragma unroll
      for (int v = 0; v < 8; ++v) acc[rt][ct][v] = 0.f;

  for (int k0 = 0; k0 < 64; k0 += 4) {
    int kk = (k0 + lk) ^ swz;         // swizzled k offset; pair stays contiguous
    v2f a[4], bb[2];
    #pragma unroll
    for (int rt = 0; rt < 4; ++rt) {
      int row = wr*64 + rt*16 + lrow;
      a[rt] = *((const v2f*)(Asb + row*64 + kk));
    }
    #pragma unroll
    for (int ct = 0; ct < 2; ++ct) {
      int col = wc*32 + ct*16 + lrow;
      bb[ct] = *((const v2f*)(Btb + col*64 + kk));
    }
    #pragma unroll
    for (int rt = 0; rt < 4; ++rt)
      #pragma unroll
      for (int ct = 0; ct < 2; ++ct)
        acc[rt][ct] = __builtin_amdgcn_wmma_f32_16x16x4_f32(
            false, a[rt], false, bb[ct], (short)0, acc[rt][ct], false, false);
  }

  // C/D layout: VGPR v -> rows v (lanes 0-15) / v+8 (lanes 16-31); col = lane&15.
  // Output is written once and never re-read (268MB > 192MB L2): stream with NT stores.
  int half = (lane >> 4) * 8;
  #pragma unroll
  for (int rt = 0; rt < 4; ++rt) {
    #pragma unroll
    for (int ct = 0; ct < 2; ++ct) {
      int colg    = j0 + wc*32 + ct*16 + lrow;
      int rowbase = i0 + wr*64 + rt*16 + half;
      float* od = out + (size_t)b*N*N + (size_t)rowbase*N + colg;
      #pragma unroll
      for (int v = 0; v < 8; ++v)
        __builtin_nontemporal_store(acc[rt][ct][v], od + (size_t)v*N);
    }
  }
}

extern "C" void kernel_launch(void* const* d_in, const int* in_sizes, int n_in,
                              void* d_out, int out_size, void* d_ws, size_t ws_size,
                              hipStream_t stream) {
  const float* Q  = (const float*)d_in[0];
  const float* K  = (const float*)d_in[1];
  const int* idxq = (const int*)d_in[2];
  const int* idxk = (const int*)d_in[3];
  float* out = (float*)d_out;
  float* ws  = (float*)d_ws;

  float* QT   = ws + OFF_QT;
  float* KT   = ws + OFF_KT;
  float* K2   = ws + OFF_K2;
  float* CS   = ws + OFF_CS;
  float* PINV = ws + OFF_PINV;
  float* T1   = ws + OFF_T1;
  float* K3   = ws + OFF_K3;

  gather_kernel<<<dim3(M, B), 64, 0, stream>>>(Q, K, idxq, idxk, QT, KT);
  k2_kernel<<<dim3(B), 64, 0, stream>>>(QT, KT, K2, CS);
  inv_kernel<<<dim3(B), 256, 0, stream>>>(K2, CS, PINV);
  k1t1_kernel<<<dim3(N/4, B), 256, 0, stream>>>(Q, KT, PINV, T1);
  k3_kernel<<<dim3(M, B), 256, 0, stream>>>(QT, K, K3);
  gemm_wmma_kernel<<<dim3(N/128, N/128, B), 256, 0, stream>>>(T1, K3, out);
}